// CLoss_83219286327664
// MI455X (gfx1250) — compile-verified
//
#include <hip/hip_runtime.h>
#include <hip/hip_bf16.h>
#include <stdint.h>

#define NROWS   4096
#define NCOLS   50257
#define H0      25128                 // half-0 length (multiple of 4 so both halves share alignment phase row&3)
#define H1      (NCOLS - H0)          // 25129
#define HCHUNKS 6283                  // ceil((3 + 25129)/4) : float4 chunks staged per half (both halves)
#define HSMEM   (HCHUNKS * 4)         // 25132 floats = 100528 B  -> 2-3 blocks per 320KB WGP
#define BLK1    512                   // 16 waves
#define BLK2    1024

// Builtin signature (from hipcc diagnostic): (v4i AS1*, v4i AS3*, imm offset, imm cpol)
typedef int v4i __attribute__((vector_size(16)));
typedef __attribute__((address_space(1))) v4i as1_v4i;
typedef __attribute__((address_space(3))) v4i as3_v4i;

#if defined(__gfx1250__) && __has_builtin(__builtin_amdgcn_global_load_async_to_lds_b128)
#define ASYNC_LDS 1
#else
#define ASYNC_LDS 0
#endif

__device__ __forceinline__ void stage16(const float* g, float* l) {
#if ASYNC_LDS
    // gfx1250 async DMA: copies 16B/lane from global address g to LDS address l, tracked by ASYNCcnt.
    __builtin_amdgcn_global_load_async_to_lds_b128(
        (as1_v4i*)(unsigned long long)g,
        (as3_v4i*)(unsigned int)(unsigned long long)l,
        0, 0);
#else
    *(float4*)l = *(const float4*)g;
#endif
}

__device__ __forceinline__ void async_wait_all() {
#if ASYNC_LDS
#if __has_builtin(__builtin_amdgcn_s_wait_asynccnt)
    __builtin_amdgcn_s_wait_asynccnt(0);
#else
    asm volatile("s_wait_asynccnt 0" ::: "memory");
#endif
#endif
}

// wave32 reductions (CDNA5 is wave32-only)
__device__ __forceinline__ float wave_max(float v) {
    for (int off = 16; off > 0; off >>= 1) v = fmaxf(v, __shfl_xor(v, off, 32));
    return v;
}
__device__ __forceinline__ float wave_sum(float v) {
    for (int off = 16; off > 0; off >>= 1) v += __shfl_xor(v, off, 32);
    return v;
}

// ---------------------------------------------------------------------------
// Kernel 1: one block per HALF-row (grid = 2*NROWS). Async-stage the half into
// LDS, neutralize pad + target slots to -inf, then two branch-free float4
// passes: local max (== masked local max) and local sum-exp.
// ---------------------------------------------------------------------------
__global__ __launch_bounds__(BLK1) void half_stats_kernel(const float* __restrict__ out,
                                                          const int*   __restrict__ target,
                                                          float*       __restrict__ hmax,
                                                          float*       __restrict__ hsum,
                                                          float*       __restrict__ tlogit)
{
    __shared__ __align__(16) float smem[HSMEM];
    __shared__ float red[16];
    __shared__ float sM, sS;

    const int tid   = threadIdx.x;
    const int lane  = tid & 31;
    const int wid   = tid >> 5;
    const int bid   = blockIdx.x;
    const int row   = bid >> 1;
    const int half  = bid & 1;
    const int start = half ? H0 : 0;
    const int len   = half ? H1 : H0;
    const int p0    = row & 3;                                       // alignment phase of this half
    const float* g_base = out + (size_t)row * NCOLS + start - p0;    // 16B aligned

    // ---- stage half-row -> LDS via async B128 DMA ----
    for (int i = tid; i < HCHUNKS; i += BLK1)
        stage16(g_base + 4 * i, &smem[4 * i]);
    async_wait_all();
    __syncthreads();

    // ---- neutralize pads and (if present) the target slot ----
    if (tid == 0) {
        const float NEG = -__builtin_inff();
        for (int k = 0; k < p0; ++k) smem[k] = NEG;                  // head pad (<=3)
        for (int k = p0 + len; k < HSMEM; ++k) smem[k] = NEG;        // tail pad (<=4)
        const int tgt = target[row];
        if (tgt >= start && tgt < start + len) {
            int tkl = p0 + (tgt - start);
            tlogit[row] = smem[tkl];                                 // exactly one half owns this
            smem[tkl]   = NEG;                                       // excluded from max; exp() -> 0
        }
    }
    __syncthreads();

    const float4* sm4 = (const float4*)smem;

    // ---- pass A: local masked max (vectorized, branch-free) ----
    float mx = -__builtin_inff();
    for (int c = tid; c < HCHUNKS; c += BLK1) {
        float4 v = sm4[c];
        mx = fmaxf(mx, fmaxf(fmaxf(v.x, v.y), fmaxf(v.z, v.w)));
    }
    mx = wave_max(mx);
    if (lane == 0) red[wid] = mx;
    __syncthreads();
    if (tid < 32) {
        float v = (tid < 16) ? red[tid] : -__builtin_inff();
        v = wave_max(v);
        if (tid == 0) sM = v;
    }
    __syncthreads();
    const float Lm = sM;

    // ---- pass B: local sum of exp (pads/target contribute exp(-inf)=0) ----
    const float LOG2E = 1.4426950408889634f;
    float s = 0.f;
    for (int c = tid; c < HCHUNKS; c += BLK1) {
        float4 v = sm4[c];
        s += exp2f((v.x - Lm) * LOG2E) + exp2f((v.y - Lm) * LOG2E)
           + exp2f((v.z - Lm) * LOG2E) + exp2f((v.w - Lm) * LOG2E);
    }
    s = wave_sum(s);
    if (lane == 0) red[wid] = s;
    __syncthreads();
    if (tid < 32) {
        float v = (tid < 16) ? red[tid] : 0.f;
        v = wave_sum(v);
        if (tid == 0) sS = v;
    }
    __syncthreads();

    if (tid == 0) {
        hmax[bid] = Lm;
        hsum[bid] = sS;
    }
}

// ---------------------------------------------------------------------------
// Kernel 2: single block. Combine half statistics (exact logsumexp merge),
// compute margin + soft-hinge l, bitonic sort (margin,id), inclusive scan,
// keep-mask, c1/c2 reductions, scalar min.
// ---------------------------------------------------------------------------
__global__ __launch_bounds__(BLK2) void finalize_kernel(const float* __restrict__ hmax,
                                                        const float* __restrict__ hsum,
                                                        const float* __restrict__ tlogit,
                                                        const int*   __restrict__ thr_p,
                                                        float*       __restrict__ out)
{
    __shared__ float key[NROWS];
    __shared__ int   sid[NROWS];
    __shared__ float cum[NROWS];
    __shared__ float lsh[NROWS];
    __shared__ float r1[32], r2[32], r3[32];

    const int tid  = threadIdx.x;
    const int lane = tid & 31;
    const int wid  = tid >> 5;
    const float LOG2E = 1.4426950408889634f;

    // ---- combine halves -> margin, l ----
    float negf = 0.f;
    for (int i = tid; i < NROWS; i += BLK2) {
        float m0 = hmax[2 * i],     m1 = hmax[2 * i + 1];
        float s0 = hsum[2 * i],     s1 = hsum[2 * i + 1];
        float tl = tlogit[i];
        float Mm = fmaxf(m0, m1);                 // row max excluding target
        float Mf = fmaxf(Mm, tl);                 // full row max
        float S  = s0 * exp2f((m0 - Mf) * LOG2E)
                 + s1 * exp2f((m1 - Mf) * LOG2E)
                 + exp2f((tl - Mf) * LOG2E);      // add target's exp back
        float lse  = Mf + __logf(S);
        float marg = tl - Mm;
        float l    = (marg > 0.f) ? (1.f - marg) : (1.f - tl + lse);
        key[i] = marg;
        sid[i] = i;
        lsh[i] = fmaxf(l, 0.f);
        if (marg < 0.f) negf += 1.f;
    }
    __syncthreads();

    // ---- bitonic sort ascending, keys with companion ids ----
    for (int k = 2; k <= NROWS; k <<= 1) {
        for (int j = k >> 1; j > 0; j >>= 1) {
            for (int i = tid; i < NROWS; i += BLK2) {
                int ixj = i ^ j;
                if (ixj > i) {
                    bool  up = ((i & k) == 0);
                    float a = key[i], b = key[ixj];
                    bool  sw = up ? (a > b) : (a < b);
                    if (sw) {
                        key[i] = b; key[ixj] = a;
                        int t = sid[i]; sid[i] = sid[ixj]; sid[ixj] = t;
                    }
                }
            }
            __syncthreads();
        }
    }

    // ---- inclusive scan of sorted margins (Hillis-Steele, 4 items/thread) ----
    for (int i = tid; i < NROWS; i += BLK2) cum[i] = key[i];
    __syncthreads();
    for (int off = 1; off < NROWS; off <<= 1) {
        float t[4];
        for (int r = 0; r < 4; ++r) {
            int i = tid + r * BLK2;
            t[r] = (i >= off) ? cum[i - off] : 0.f;
        }
        __syncthreads();
        for (int r = 0; r < 4; ++r) cum[tid + r * BLK2] += t[r];
        __syncthreads();
    }

    // ---- keep[i] = cum[i] <= threshold + 1 - i ; c1 = sum keep * l[sorted id] ----
    const float thr = (float)thr_p[0];
    float c1 = 0.f, vsum = 0.f;
    for (int r = 0; r < 4; ++r) {
        int i = tid + r * BLK2;
        if (cum[i] <= thr + 1.0f - (float)i) {
            c1   += lsh[sid[i]];
            vsum += 1.f;
        }
    }

    c1 = wave_sum(c1); vsum = wave_sum(vsum); negf = wave_sum(negf);
    if (lane == 0) { r1[wid] = c1; r2[wid] = vsum; r3[wid] = negf; }
    __syncthreads();
    if (tid < 32) {
        c1   = wave_sum(r1[tid]);
        vsum = wave_sum(r2[tid]);
        negf = wave_sum(r3[tid]);
        if (tid == 0) {
            float c2 = (float)NROWS - vsum + negf;
            out[0] = (c1 < c2) ? c1 : c2;
        }
    }
}

extern "C" void kernel_launch(void* const* d_in, const int* in_sizes, int n_in,
                              void* d_out, int out_size, void* d_ws, size_t ws_size,
                              hipStream_t stream) {
    (void)in_sizes; (void)n_in; (void)out_size; (void)ws_size;
    const float* output = (const float*)d_in[0];
    const int*   target = (const int*)d_in[1];
    const int*   thresh = (const int*)d_in[2];
    float* hmax   = (float*)d_ws;          // [2*NROWS]
    float* hsum   = hmax + 2 * NROWS;      // [2*NROWS]
    float* tlogit = hsum + 2 * NROWS;      // [NROWS]

    half_stats_kernel<<<2 * NROWS, BLK1, 0, stream>>>(output, target, hmax, hsum, tlogit);
    finalize_kernel<<<1, BLK2, 0, stream>>>(hmax, hsum, tlogit, thresh, (float*)d_out);
}